// LightGCN_53120155517443
// MI455X (gfx1250) — compile-verified
//
#include <hip/hip_runtime.h>
#include <hip/hip_bf16.h>
#include <stdint.h>

typedef __attribute__((ext_vector_type(2))) float v2f;
typedef __attribute__((ext_vector_type(8))) float v8f;

#define NUM_USERS 200000
#define NUM_ITEMS 100000
#define HIDD 64
#define NLAYERS 3
#define NNZ_E 4800000
#define IMG_D 512
#define TXT_D 384
#define CAT_D 896
#define MID_D 256
#define HIST_L 20
#define N_NODES 300000

// GEMM tiling
#define BM 160   // 10 waves x 16 rows
#define BN 64    // 4 wmma N-tiles
#define BK 32    // K-slice per stage
#define TPB 320  // 10 waves

// ---------------------------------------------------------------------------
// CDNA5 async global->LDS copy (ASYNCcnt) + wait, via inline asm.
// ---------------------------------------------------------------------------
__device__ __forceinline__ uint32_t lds_off(const void* p) {
  // low 32 bits of a flat shared-aperture address == LDS byte offset
  return (uint32_t)(uintptr_t)p;
}
__device__ __forceinline__ void async_cp_b128(uint32_t ldsoff, const void* g) {
  asm volatile("global_load_async_to_lds_b128 %0, %1, off"
               :: "v"(ldsoff), "v"(g)
               : "memory");
}
__device__ __forceinline__ void wait_async_barrier() {
  asm volatile("s_wait_asynccnt 0" ::: "memory");
  __syncthreads();
}

// XOR swizzle on float4 chunks: row r, chunk c (0..7) -> slot c ^ (r & 7).
// Keeps 16B alignment for b128 async writes; spreads wmma-operand ds reads
// (lane stride = 32 floats) across banks (<=2-way conflict instead of 8-way).
__device__ __forceinline__ float lds_read(const float* base, int r, int k) {
  const int c = ((k >> 2) ^ (r & 7));
  return base[r * BK + c * 4 + (k & 3)];
}

// ---------------------------------------------------------------------------
// Tiled fp32 WMMA GEMM:  C[M,N] = A[M,K] @ W[N,K]^T
// A is a virtual row-major concat: cols [0,kb) from A0 (ld0), [kb,K) from A1
// (ld1). kb is a multiple of BK so each K-tile lives wholly in one tensor.
// Block = 320 threads / 10 waves -> 160x64 tile; double-buffered async LDS
// staging; operand fetch software-pipelined one k-step ahead of the WMMAs.
// ---------------------------------------------------------------------------
__global__ __launch_bounds__(TPB)
void gemm_wmma_tiled(const float* __restrict__ A0, int ld0,
                     const float* __restrict__ A1, int ld1, int kb,
                     const float* __restrict__ W, float* __restrict__ C,
                     int N, int K) {
  __shared__ float lA[2][BM * BK];
  __shared__ float lB[2][BN * BK];

  const int tid  = threadIdx.x;
  const int lane = tid & 31;
  const int wv   = tid >> 5;        // 0..9
  const int half = lane >> 4;
  const int nl   = lane & 15;
  const int row0 = blockIdx.x * BM;
  const int n0   = blockIdx.y * BN;
  const int wrow = wv * 16;         // wave's row offset inside the block tile

  auto stage = [&](int buf, int k0) {
    const float* abase;
    int ld;
    if (k0 < kb) { abase = A0 + k0; ld = ld0; }
    else         { abase = A1 + (k0 - kb); ld = ld1; }
    // A tile: BM*8 = 1280 float4 chunks, 4 per thread
    for (int idx = tid; idx < BM * 8; idx += TPB) {
      const int r = idx >> 3, c = idx & 7;
      const int slot = r * 8 + (c ^ (r & 7));
      async_cp_b128(lds_off(&lA[buf][slot * 4]),
                    abase + (size_t)(row0 + r) * ld + 4 * c);
    }
    // B tile: BN*8 = 512 float4 chunks
    for (int idx = tid; idx < BN * 8; idx += TPB) {
      const int r = idx >> 3, c = idx & 7;
      const int slot = r * 8 + (c ^ (r & 7));
      async_cp_b128(lds_off(&lB[buf][slot * 4]),
                    W + (size_t)(n0 + r) * K + k0 + 4 * c);
    }
  };

  v8f acc[4] = {};

  stage(0, 0);
  wait_async_barrier();

  const int rA = wrow + nl;
  const int nt = K / BK;
  for (int t = 0; t < nt; ++t) {
    const int cur = t & 1;
    if (t + 1 < nt) stage(cur ^ 1, (t + 1) * BK);   // overlaps compute below

    const float* pa = lA[cur];
    const float* pb = lB[cur];

    // -- software-pipelined operand fetch: load step kk+4 during wmma(kk) --
    v2f a_cur, b_cur[4];
    {
      const int ka = 2 * half;
      a_cur.x = lds_read(pa, rA, ka);
      a_cur.y = lds_read(pa, rA, ka + 1);
#pragma unroll
      for (int j = 0; j < 4; ++j) {
        const int rB = 16 * j + nl;
        b_cur[j].x = lds_read(pb, rB, ka);
        b_cur[j].y = lds_read(pb, rB, ka + 1);
      }
    }
#pragma unroll
    for (int kk = 0; kk < BK; kk += 4) {
      v2f a_nxt, b_nxt[4];
      if (kk + 4 < BK) {
        const int ka = kk + 4 + 2 * half;
        a_nxt.x = lds_read(pa, rA, ka);
        a_nxt.y = lds_read(pa, rA, ka + 1);
#pragma unroll
        for (int j = 0; j < 4; ++j) {
          const int rB = 16 * j + nl;
          b_nxt[j].x = lds_read(pb, rB, ka);
          b_nxt[j].y = lds_read(pb, rB, ka + 1);
        }
      }
#pragma unroll
      for (int j = 0; j < 4; ++j)
        acc[j] = __builtin_amdgcn_wmma_f32_16x16x4_f32(
            false, a_cur, false, b_cur[j], (short)0, acc[j], false, false);
      if (kk + 4 < BK) {
        a_cur = a_nxt;
#pragma unroll
        for (int j = 0; j < 4; ++j) b_cur[j] = b_nxt[j];
      }
    }
    wait_async_barrier();   // next tile staged + all waves done with buf[cur]
  }

  // C/D 16x16 layout: VGPR r -> M = r + 8*half, N = lane&15
#pragma unroll
  for (int j = 0; j < 4; ++j)
#pragma unroll
    for (int r = 0; r < 8; ++r)
      C[(size_t)(row0 + wrow + r + 8 * half) * N + n0 + 16 * j + nl] = acc[j][r];
}

// ---------------------------------------------------------------------------
// Fused per-item MHA: 64 threads per item; emb_user gathers are L2-resident.
// ---------------------------------------------------------------------------
__global__ __launch_bounds__(64)
void mha_kernel(const int* __restrict__ hist, const float* __restrict__ emb_user,
                const float* __restrict__ emb_item,
                const float* __restrict__ Qw, const float* __restrict__ Kw,
                const float* __restrict__ Vw, float* __restrict__ out) {
  __shared__ float squery[HIDD];
  __shared__ float sq[HIDD];
  __shared__ float sA[HIST_L][HIDD];
  __shared__ float sk[HIST_L][HIDD];
  __shared__ float sv[HIST_L][HIDD];

  const int item = blockIdx.x;
  const int t = threadIdx.x;

  squery[t] = emb_item[(size_t)item * HIDD + t];
  for (int l = 0; l < HIST_L; ++l) {
    const int u = hist[(size_t)item * HIST_L + l];
    sA[l][t] = emb_user[(size_t)u * HIDD + t];
  }
  __syncthreads();

  {
    float a = 0.f;
    for (int d = 0; d < HIDD; ++d) a += squery[d] * Qw[d * HIDD + t];
    sq[t] = a;
  }
  for (int l = 0; l < HIST_L; ++l) {
    float ak = 0.f, av = 0.f;
    for (int d = 0; d < HIDD; ++d) {
      const float x = sA[l][d];
      ak += x * Kw[d * HIDD + t];
      av += x * Vw[d * HIDD + t];
    }
    sk[l][t] = ak;
    sv[l][t] = av;
  }
  __syncthreads();

  const int h = t >> 4;
  float s[HIST_L];
  float mx = -1e30f;
  for (int l = 0; l < HIST_L; ++l) {
    float sc = 0.f;
#pragma unroll
    for (int a = 0; a < 16; ++a) sc += sq[h * 16 + a] * sk[l][h * 16 + a];
    sc *= 0.125f;
    s[l] = sc;
    mx = fmaxf(mx, sc);
  }
  float sum = 0.f;
  for (int l = 0; l < HIST_L; ++l) {
    s[l] = __expf(s[l] - mx);
    sum += s[l];
  }
  float r = 0.f;
  for (int l = 0; l < HIST_L; ++l) r += s[l] * sv[l][t];
  out[(size_t)item * HIDD + t] = r / sum;
}

// ---------------------------------------------------------------------------
// all_emb = [emb_user ; emb_item * fea + one_hop]  -> acc and x0
// ---------------------------------------------------------------------------
__global__ void combine_kernel(const float* __restrict__ emb_user,
                               const float* __restrict__ emb_item,
                               const float* __restrict__ fea,
                               const float* __restrict__ onehop,
                               float* __restrict__ acc, float* __restrict__ x) {
  const size_t i = (size_t)blockIdx.x * blockDim.x + threadIdx.x;
  if (i >= (size_t)N_NODES * HIDD) return;
  float v;
  if (i < (size_t)NUM_USERS * HIDD) {
    v = emb_user[i];
  } else {
    const size_t j = i - (size_t)NUM_USERS * HIDD;
    v = emb_item[j] * fea[j] + onehop[j];
  }
  acc[i] = v;
  x[i] = v;
}

__global__ void zero_kernel(float4* __restrict__ p, size_t n4) {
  const size_t i = (size_t)blockIdx.x * blockDim.x + threadIdx.x;
  if (i < n4) p[i] = make_float4(0.f, 0.f, 0.f, 0.f);
}

// Edge-parallel SpMM: y[row] += val * x[col]; 16 lanes/edge, float4/lane.
// x table (76.8 MB) fits in the 192 MB L2 -> gathers + atomics stay on-chip.
__global__ __launch_bounds__(256)
void spmm_kernel(const int* __restrict__ rows, const int* __restrict__ cols,
                 const float* __restrict__ vals, const float* __restrict__ x,
                 float* __restrict__ y) {
  const size_t t = (size_t)blockIdx.x * blockDim.x + threadIdx.x;
  const size_t e = t >> 4;
  const int j = (int)(t & 15);
  if (e >= (size_t)NNZ_E) return;
  const int r = rows[e];
  const int c = cols[e];
  const float val = vals[e];
  const float4 xv = *(const float4*)(x + (size_t)c * HIDD + 4 * j);
  float* yp = y + (size_t)r * HIDD + 4 * j;
  atomicAdd(yp + 0, val * xv.x);
  atomicAdd(yp + 1, val * xv.y);
  atomicAdd(yp + 2, val * xv.z);
  atomicAdd(yp + 3, val * xv.w);
}

__global__ void accadd_kernel(float4* __restrict__ acc, const float4* __restrict__ x,
                              size_t n4) {
  const size_t i = (size_t)blockIdx.x * blockDim.x + threadIdx.x;
  if (i < n4) {
    float4 a = acc[i];
    const float4 b = x[i];
    a.x += b.x; a.y += b.y; a.z += b.z; a.w += b.w;
    acc[i] = a;
  }
}

__global__ void scale_kernel(float4* __restrict__ p, float s, size_t n4) {
  const size_t i = (size_t)blockIdx.x * blockDim.x + threadIdx.x;
  if (i < n4) {
    float4 a = p[i];
    a.x *= s; a.y *= s; a.z *= s; a.w *= s;
    p[i] = a;
  }
}

// ---------------------------------------------------------------------------
extern "C" void kernel_launch(void* const* d_in, const int* in_sizes, int n_in,
                              void* d_out, int out_size, void* d_ws, size_t ws_size,
                              hipStream_t stream) {
  const int*   hist      = (const int*)d_in[0];
  const int*   grows     = (const int*)d_in[1];
  const int*   gcols     = (const int*)d_in[2];
  const float* gvals     = (const float*)d_in[3];
  const float* emb_user  = (const float*)d_in[4];
  const float* emb_item  = (const float*)d_in[5];
  const float* img       = (const float*)d_in[6];
  const float* txt       = (const float*)d_in[7];
  const float* Qw        = (const float*)d_in[8];
  const float* Kw        = (const float*)d_in[9];
  const float* Vw        = (const float*)d_in[10];
  const float* W_onehop  = (const float*)d_in[11];
  const float* W_mul1    = (const float*)d_in[12];
  const float* W_mul2    = (const float*)d_in[13];

  float* ws  = (float*)d_ws;
  float* acc = (float*)d_out;   // [N_NODES, 64] running layer sum

  // workspace layout (aliased across phases):
  //   [0, I*256)             : C1 (dense phase)  |  xA (SpMM phase)
  //   [I*256, I*256+3*I*64)  : fea, mha, onehop  |  xB (SpMM phase)
  float* c1     = ws;
  float* fea    = ws + (size_t)NUM_ITEMS * MID_D;
  float* mhab   = fea + (size_t)NUM_ITEMS * HIDD;
  float* onehop = mhab + (size_t)NUM_ITEMS * HIDD;
  float* xA     = ws;
  float* xB     = ws + (size_t)NUM_ITEMS * MID_D;

  // 1) C1 = concat(img,txt) @ W_mul1^T   [100000, 256]
  {
    dim3 grid(NUM_ITEMS / BM, MID_D / BN);
    gemm_wmma_tiled<<<grid, TPB, 0, stream>>>(img, IMG_D, txt, TXT_D, IMG_D,
                                              W_mul1, c1, MID_D, CAT_D);
  }
  // 2) fea = C1 @ W_mul2^T   [100000, 64]
  {
    dim3 grid(NUM_ITEMS / BM, 1);
    gemm_wmma_tiled<<<grid, TPB, 0, stream>>>(c1, MID_D, c1, MID_D, MID_D,
                                              W_mul2, fea, HIDD, MID_D);
  }
  // 3) mha_out
  mha_kernel<<<NUM_ITEMS, 64, 0, stream>>>(hist, emb_user, emb_item, Qw, Kw, Vw, mhab);
  // 4) one_hop = mha_out @ W_onehop^T   [100000, 64]
  {
    dim3 grid(NUM_ITEMS / BM, 1);
    gemm_wmma_tiled<<<grid, TPB, 0, stream>>>(mhab, HIDD, mhab, HIDD, HIDD,
                                              W_onehop, onehop, HIDD, HIDD);
  }
  // 5) all_emb -> acc (=d_out) and xA
  {
    const size_t total = (size_t)N_NODES * HIDD;
    combine_kernel<<<(unsigned)((total + 255) / 256), 256, 0, stream>>>(
        emb_user, emb_item, fea, onehop, acc, xA);
  }
  // 6) 3 propagation layers: x' = segsum(x[cols]*vals -> rows); acc += x'
  {
    const size_t total = (size_t)N_NODES * HIDD;
    const size_t n4 = total / 4;
    float* cur = xA;
    float* nxt = xB;
    for (int layer = 0; layer < NLAYERS; ++layer) {
      zero_kernel<<<(unsigned)((n4 + 255) / 256), 256, 0, stream>>>((float4*)nxt, n4);
      const size_t spmm_threads = (size_t)NNZ_E * 16;
      spmm_kernel<<<(unsigned)((spmm_threads + 255) / 256), 256, 0, stream>>>(
          grows, gcols, gvals, cur, nxt);
      accadd_kernel<<<(unsigned)((total / 4 + 255) / 256), 256, 0, stream>>>(
          (float4*)acc, (const float4*)nxt, n4);
      float* tmp = cur; cur = nxt; nxt = tmp;
    }
    // 7) light = acc / (N_LAYERS + 1)
    scale_kernel<<<(unsigned)((n4 + 255) / 256), 256, 0, stream>>>((float4*)acc, 0.25f, n4);
  }
}